// MPS_tensor_40638980555111
// MI455X (gfx1250) — compile-verified
//
#include <hip/hip_runtime.h>

// ---------------------------------------------------------------------------
// MPS chain contraction for MI455X (gfx1250, wave32, WMMA).
//
// Per sweep step:  y_new[n,h,r] = sum_{c,l,d} y[n,c,l]*x_t[n,c,d]*T_t[c,h,l,d,r]
// mapped to WMMA_F32_16x16x4_F32:  M=n-tile(16), N=(h,r) 8 tiles of 16,
// K=(c,l,d)=256 -> 64 chained fp32 WMMAs per wave per step.
// Bond state lives in LDS between steps; core slab streamed from L2.
// ---------------------------------------------------------------------------

typedef __attribute__((ext_vector_type(2))) float v2f;
typedef __attribute__((ext_vector_type(8))) float v8f;

#define N_TOTAL 256
#define C_IN    8
#define C_OUT   8
#define DIN     2
#define LENG    1024
#define BOND    16
#define POS_OUT 512

// tensors strides in floats: [c][h][t][l][d][r]
#define TC  (C_OUT*LENG*BOND*DIN*BOND)   // 4194304
#define TH  (LENG*BOND*DIN*BOND)         // 524288
#define TT  (BOND*DIN*BOND)              // 512
#define TL  (DIN*BOND)                   // 32
#define TD  (BOND)                       // 16

// x strides in floats: [n][c][d][1][t]
#define XN (C_IN*DIN*LENG)               // 16384
#define XC (DIN*LENG)                    // 2048
#define XD (LENG)                        // 1024

#define MT 16          // n rows per block (one WMMA M tile)
#define YL_STRIDE 129  // padded row stride for bond state (bank spread)
#define XB_STRIDE 17   // padded row stride for x_t

__global__ __launch_bounds__(256) void mps_sweep_kernel(
    const float* __restrict__ x, const float* __restrict__ T,
    float* __restrict__ ws)
{
    __shared__ float ylbuf[MT * YL_STRIDE];   // bond state y[n][c][16]
    __shared__ float xbuf [MT * XB_STRIDE];   // x_t[n][c][d]

    const int tid   = threadIdx.x;
    const int lane  = tid & 31;
    const int wave  = tid >> 5;               // h-tile owned by this wave
    const int nbase = blockIdx.x * MT;
    const bool right = (blockIdx.y != 0);

    // ---- init bond state ----
    // left : yl0[n,h,r] = sum_{c,d} x[n,c,d,0]    * T[c,h,t=0,l=0,d,r]
    // right: yr0[n,h,l] = sum_{c,d} x[n,c,d,1023] * T[c,h,t=0,l,d,r=0]
    {
        const int tx = right ? (LENG - 1) : 0;
        for (int o = tid; o < MT * C_OUT * BOND; o += 256) {
            const int n = o >> 7;
            const int h = (o >> 4) & 7;
            const int b = o & 15;             // r (left) or l (right)
            float acc = 0.f;
            for (int c = 0; c < C_IN; ++c)
                for (int d = 0; d < DIN; ++d) {
                    const float xv = x[(nbase + n) * XN + c * XC + d * XD + tx];
                    const float tv = right ? T[c * TC + h * TH + b * TL + d * TD]
                                           : T[c * TC + h * TH + d * TD + b];
                    acc += xv * tv;
                }
            ylbuf[n * YL_STRIDE + h * BOND + b] = acc;
        }
    }

    const int nsteps = right ? (LENG - 2 - POS_OUT) : POS_OUT;   // 510 : 512

    const int row   = lane & 15;
    const int col   = lane & 15;
    const int khalf = (lane >> 4) << 1;       // 0 (lanes 0-15) or 2 (16-31)
    const int rbase = (lane >> 4) << 3;       // C/D row base: 0 or 8

    for (int i = 0; i < nsteps; ++i) {
        const int tx = right ? (LENG - 1 - i) : i;        // x site
        const int tt = right ? (LENG - 2 - i) : (i + 1);  // tensor site

        // stage x_t (one float per thread: n=tid>>4, cd=tid&15)
        {
            const int n = tid >> 4, cd = tid & 15;
            const int c = cd >> 1, d = cd & 1;
            xbuf[n * XB_STRIDE + cd] = x[(nbase + n) * XN + c * XC + d * XD + tx];
        }
        __syncthreads();   // prev ylbuf writes + xbuf now visible

        // prefetch next step's slab chunks for this wave's h (L2 resident)
        if (i + 1 < nsteps) {
            const int ttn = right ? (tt - 1) : (tt + 1);
            const float* pb = T + (size_t)ttn * TT + (size_t)wave * TH + lane * 16;
            #pragma unroll
            for (int c = 0; c < C_IN; ++c)
                __builtin_prefetch(pb + (size_t)c * TC, 0, 0);
        }

        const float* Tt   = T + (size_t)tt * TT + (size_t)wave * TH;
        const float* yrow = ylbuf + row * YL_STRIDE;
        const float* xrow = xbuf  + row * XB_STRIDE;

        v8f acc = {};
        #pragma unroll 8
        for (int kk = 0; kk < 64; ++kk) {
            const int k0 = kk * 4 + khalf;
            const int k1 = k0 + 1;
            // k = (c<<5) | (bond<<1) | d
            const int c0 = k0 >> 5, b0 = (k0 >> 1) & 15, d0 = k0 & 1;
            const int c1 = k1 >> 5, b1 = (k1 >> 1) & 15, d1 = k1 & 1;

            v2f a, b;
            a.x = yrow[c0 * BOND + b0] * xrow[c0 * 2 + d0];
            a.y = yrow[c1 * BOND + b1] * xrow[c1 * 2 + d1];
            if (!right) {   // contract l, output bond r = col (coalesced in r)
                b.x = Tt[c0 * TC + b0 * TL + d0 * TD + col];
                b.y = Tt[c1 * TC + b1 * TL + d1 * TD + col];
            } else {        // contract r, output bond l = col
                b.x = Tt[c0 * TC + col * TL + d0 * TD + b0];
                b.y = Tt[c1 * TC + col * TL + d1 * TD + b1];
            }
            acc = __builtin_amdgcn_wmma_f32_16x16x4_f32(
                false, a, false, b, (short)0, acc, false, false);
        }

        __syncthreads();   // everyone done reading old bond state

        // D layout -> LDS: VGPR v = row rbase+v, column = col, h-tile = wave
        #pragma unroll
        for (int v = 0; v < 8; ++v)
            ylbuf[(rbase + v) * YL_STRIDE + wave * BOND + col] = acc[v];
        // next iteration's __syncthreads publishes these writes
    }
    __syncthreads();

    // final bond state -> workspace (left first half, right second half)
    float* wsout = ws + (right ? N_TOTAL * C_OUT * BOND : 0);
    for (int o = tid; o < MT * C_OUT * BOND; o += 256) {
        const int n = o >> 7, h = (o >> 4) & 7, b = o & 15;
        wsout[(size_t)(nbase + n) * C_OUT * BOND + h * BOND + b] =
            ylbuf[n * YL_STRIDE + h * BOND + b];
    }
}

// out[n,c] = (sum_r yl[n,c,r]) * (sum_l yr[n,c,l])   (einsum 'ncr,ncl->nc')
__global__ __launch_bounds__(256) void mps_combine_kernel(
    const float* __restrict__ ws, float* __restrict__ out)
{
    const int tid = blockIdx.x * 256 + threadIdx.x;
    if (tid >= N_TOTAL * C_OUT) return;
    const float* yl = ws + (size_t)tid * BOND;
    const float* yr = ws + (size_t)N_TOTAL * C_OUT * BOND + (size_t)tid * BOND;
    float sl = 0.f, sr = 0.f;
    #pragma unroll
    for (int r = 0; r < BOND; ++r) { sl += yl[r]; sr += yr[r]; }
    out[tid] = sl * sr;
}

extern "C" void kernel_launch(void* const* d_in, const int* in_sizes, int n_in,
                              void* d_out, int out_size, void* d_ws, size_t ws_size,
                              hipStream_t stream) {
    const float* x = (const float*)d_in[0];       // (256,8,2,1,1024) f32
    const float* T = (const float*)d_in[1];       // (8,8,1024,16,2,16) f32
    // d_in[2] = pos_out (scalar 512, matches POS_OUT)
    float* ws = (float*)d_ws;                     // needs 2*256*128*4 = 256 KB

    dim3 grid(N_TOTAL / MT, 2);                   // 16 n-tiles x {left,right}
    mps_sweep_kernel<<<grid, 256, 0, stream>>>(x, T, ws);
    mps_combine_kernel<<<(N_TOTAL * C_OUT + 255) / 256, 256, 0, stream>>>(
        ws, (float*)d_out);
}